// YOLO_loss_49220325212874
// MI455X (gfx1250) — compile-verified
//
#include <hip/hip_runtime.h>

typedef __attribute__((ext_vector_type(2))) float v2f;
typedef __attribute__((ext_vector_type(8))) float v8f;

#define N_CELLS   776
#define N_ANCH    3
#define N_CH      7
#define CONF_TH   0.8f
#define NO_OBJ    0.5f
#define BLOCK     256
#define NWAVES    (BLOCK / 32)

// Sum a float across the 32 lanes of a wave using the CDNA5 matrix pipe.
// A (16x4 f32): a0 -> A[m][0] = x_m (lanes 0-15), A[m][2] = x_{m+16} (lanes 16-31);
// a1 -> K=1/3 set to 0.  B = all ones (layout independent).
// D[m][n] = x_m + x_{m+16} for every n.  Lane L holds D[0..7][L] in its 8 dest
// VGPRs, lane L^16 holds D[8..15][L]; sum-of-8 + shfl_xor(16) = full wave total.
// EXEC must be all 1s at the call site.
__device__ __forceinline__ float wave_sum_wmma(float x) {
  v2f a; a[0] = x;    a[1] = 0.0f;
  v2f b; b[0] = 1.0f; b[1] = 1.0f;
  v8f c = {0.f, 0.f, 0.f, 0.f, 0.f, 0.f, 0.f, 0.f};
  v8f d = __builtin_amdgcn_wmma_f32_16x16x4_f32(
      /*neg_a=*/false, a, /*neg_b=*/false, b,
      /*c_mod=*/(short)0, c, /*reuse_a=*/false, /*reuse_b=*/false);
  float s = ((d[0] + d[1]) + (d[2] + d[3])) + ((d[4] + d[5]) + (d[6] + d[7]));
  s += __shfl_xor(s, 16, 32);
  return s;  // every lane holds the 32-lane total
}

__global__ void __launch_bounds__(BLOCK)
yolo_loss_kernel(const float* __restrict__ pred,
                 const float* __restrict__ label,
                 float* __restrict__ out) {
  const int tid = threadIdx.x;

  // label is uniform; these lower to scalar loads.
  const float lx = label[0], ly = label[1], lw = label[2], lh = label[3];
  const float lconf = label[4], lc5 = label[5], lc6 = label[6];
  const float area_l = fabsf(lw * lh);
  const float lxm = lx - 0.5f * lw, lym = ly - 0.5f * lh;
  const float lxp = lx + 0.5f * lw, lyp = ly + 0.5f * lh;
  const float sqlx = sqrtf(lx), sqly = sqrtf(ly);

  float acc = 0.0f;
  for (int cell = tid; cell < N_CELLS; cell += BLOCK) {
    const float* row = pred + cell * (N_ANCH * N_CH);
    float best = -1.0f;                 // IoU >= 0, so anchor 0 always wins first;
    float b0 = 0.f, b1 = 0.f, b4 = 0.f, b5 = 0.f, b6 = 0.f;
#pragma unroll
    for (int a = 0; a < N_ANCH; ++a) { // strict '>' == argmax first-max semantics
      const float* r = row + a * N_CH;
      const float px = r[0], py = r[1], pw = r[2], ph = r[3];
      const float ax = fmaxf(px - 0.5f * pw, lxm);
      const float ay = fmaxf(py - 0.5f * ph, lym);
      const float bx = fminf(px + 0.5f * pw, lxp);
      const float by = fminf(py + 0.5f * ph, lyp);
      const float inter = fabsf(fmaxf(bx - ax, 0.f) * fmaxf(by - ay, 0.f));
      const float iou = inter / (fabsf(pw * ph) + area_l - inter);
      if (iou > best) { best = iou; b0 = px; b1 = py; b4 = r[4]; b5 = r[5]; b6 = r[6]; }
    }
    // xy_loss + wh_loss (reference reuses channels 0/1 inside the sqrt term)
    const float dx = lx - b0, dy = ly - b1;
    const float sx = sqlx - sqrtf(b0), sy = sqly - sqrtf(b1);
    float loss = dx * dx + dy * dy + sx * sx + sy * sy;
    float csq = lconf - b4; csq *= csq;
    const float d5 = lc5 - b5, d6 = lc6 - b6;
    loss += (b4 > CONF_TH) ? (d5 * d5 + d6 * d6 + csq) : (NO_OBJ * csq);
    acc += loss;
  }

  // All 256 threads reconverged here -> EXEC all-ones for the WMMA.
  const float wsum = wave_sum_wmma(acc);

  __shared__ float s_wave[NWAVES];
  const int lane = tid & 31;
  const int wave = tid >> 5;
  if (lane == 0) s_wave[wave] = wsum;
  __syncthreads();

  if (tid < 32) {  // full wave 0 -> EXEC all-ones within this branch
    const float v = (lane < NWAVES) ? s_wave[lane] : 0.0f;
    const float total = wave_sum_wmma(v);
    if (lane == 0) out[0] = total;
  }
}

extern "C" void kernel_launch(void* const* d_in, const int* in_sizes, int n_in,
                              void* d_out, int out_size, void* d_ws, size_t ws_size,
                              hipStream_t stream) {
  const float* pred  = (const float*)d_in[0];   // (776*3, 7) f32
  const float* label = (const float*)d_in[1];   // (7,) f32
  float* out = (float*)d_out;                   // scalar f32
  (void)in_sizes; (void)n_in; (void)out_size; (void)d_ws; (void)ws_size;
  yolo_loss_kernel<<<1, BLOCK, 0, stream>>>(pred, label, out);
}